// get_3d_points_18425409700385
// MI455X (gfx1250) — compile-verified
//
#include <hip/hip_runtime.h>

// ---------------------------------------------------------------------------
// get_3d_points for MI455X (gfx1250, wave32)
//
// Shapes (fixed by the reference): B=4, C=32, H=128, W=240, J=15, HW=30720.
// Output [J,B,HW,35] f32 = 258 MB  ->  pure bandwidth problem (23.3 TB/s).
//
// Pipeline (all on `stream`):
//   1) setup_kernel  : per-batch Kinv params + T = trans @ inv(M) (analytic
//                      rigid inverse) -> 16 floats/batch in d_ws.
//   2) point_kernel  : per-pixel undistortion (f32 VALU) + point transform
//                      done with V_WMMA_F32_16X16X4_F32 (D = T(16x4) x
//                      depth_cam(4x16), rows>=3 zero), plus feature
//                      transpose into staging pc[B][HW][35] (17 MB, L2-res).
//   3) broadcast_kernel: out[j,b,p,k] = mask[j,b,p] ? pc[b,p,k] : 0,
//                      float4 NON-TEMPORAL stores (258 MB write-once stream,
//                      keep pc+mask L2-resident), fully coalesced.
// ---------------------------------------------------------------------------

typedef float v2f __attribute__((ext_vector_type(2)));
typedef float v4f __attribute__((ext_vector_type(4)));
typedef float v8f __attribute__((ext_vector_type(8)));

#define B_   4
#define C_   32
#define H_   128
#define W_   240
#define J_   15
#define HW_  (H_ * W_)       // 30720
#define NF   (3 + C_)        // 35 floats per point

// consts layout per batch (16 floats): [ifx, ify, -cx/fx, -cy/fy, T00..T03, T10..T13, T20..T23]
__global__ void setup_kernel(const float* __restrict__ K,
                             const float* __restrict__ M,
                             const float* __restrict__ trans,
                             float* __restrict__ consts) {
    int b = threadIdx.x;
    if (b >= B_) return;
    const float* Kb = K + b * 9;
    float fx = Kb[0], cx = Kb[2], fy = Kb[4], cy = Kb[5];
    float ifx = 1.0f / fx, ify = 1.0f / fy;
    float* cb = consts + b * 16;
    cb[0] = ifx; cb[1] = ify; cb[2] = -cx * ifx; cb[3] = -cy * ify;

    // inv(M) for rigid M = [R t; 0 1]  ->  [R^T, -R^T t; 0 1]
    const float* Mb = M + b * 16;
    float Minv[4][4];
#pragma unroll
    for (int i = 0; i < 3; ++i)
#pragma unroll
        for (int j = 0; j < 3; ++j)
            Minv[i][j] = Mb[j * 4 + i];
#pragma unroll
    for (int i = 0; i < 3; ++i)
        Minv[i][3] = -(Mb[0 * 4 + i] * Mb[3] + Mb[1 * 4 + i] * Mb[7] + Mb[2 * 4 + i] * Mb[11]);
    Minv[3][0] = 0.f; Minv[3][1] = 0.f; Minv[3][2] = 0.f; Minv[3][3] = 1.f;

    // T = trans @ inv(M), keep rows 0..2
    const float* Tr = trans + b * 16;
#pragma unroll
    for (int i = 0; i < 3; ++i)
#pragma unroll
        for (int j = 0; j < 4; ++j) {
            float s = 0.f;
#pragma unroll
            for (int k = 0; k < 4; ++k) s += Tr[i * 4 + k] * Minv[k][j];
            cb[4 + i * 4 + j] = s;
        }
}

__global__ __launch_bounds__(256)
void point_kernel(const float* __restrict__ depth,
                  const float* __restrict__ diff,
                  const float* __restrict__ feat,
                  const float* __restrict__ consts,
                  float* __restrict__ pc) {
    const int blocksPerBatch = HW_ / 256;                 // 120
    int b     = blockIdx.x / blocksPerBatch;
    int pBase = (blockIdx.x % blocksPerBatch) * 256;
    int p     = pBase + threadIdx.x;
    int lane  = threadIdx.x & 31;

    const float* cb = consts + b * 16;
    float ifx = cb[0], ify = cb[1], pcx = cb[2], pcy = cb[3];
    const float* T = cb + 4;                              // 3x4 row-major
    const float* db = diff + b * 12;
    float d0 = db[0], d1 = db[1], d2 = db[2],  d3 = db[3];
    float d4 = db[4], d5 = db[5], d6 = db[6],  d7 = db[7];
    float d8 = db[8], d9 = db[9], d10 = db[10], d11 = db[11];

    // pixel grid: x = (col+71)*(1920/480), y = row*(1080/128)
    int xi = p % W_;
    int yi = p / W_;
    float xcor = (float)(xi + 71) * 4.0f;
    float ycor = (float)yi * (1080.0f / 128.0f);
    float xb = xcor * ifx + pcx;                          // Kinv row 0
    float yb = ycor * ify + pcy;                          // Kinv row 1

    // 5 fixed undistortion iterations (f32, matches reference)
    float xd = xb, yd = yb;
#pragma unroll
    for (int it = 0; it < 5; ++it) {
        float r2  = xd * xd + yd * yd;
        float num = 1.0f + ((d7 * r2 + d6) * r2 + d5) * r2;
        float den = 1.0f + ((d4 * r2 + d1) * r2 + d0) * r2;
        float ic  = num / den;
        float dX  = 2.0f * d2 * xd * yd + d3 * (r2 + 2.0f * xd * xd) + d8 * r2 + d9 * r2 * r2;
        float dY  = d2 * (r2 + 2.0f * yd * yd) + 2.0f * d3 * xd * yd + d10 * r2 + d11 * r2 * r2;
        xd = (xb - dX) * ic;
        yd = (yb - dY) * ic;
    }

    float dep = depth[b * HW_ + p];
    float xc = xd * dep, yc = yd * dep, zc = dep;         // depth_cam = [xc,yc,zc,1]

    // ---- A operand: 16x4 f32 = T rows (rows>=3 zero) ----
    // ISA layout: lanes 0-15 hold M=lane with VGPR0=K0, VGPR1=K1;
    //             lanes 16-31 hold M=lane-16 with VGPR0=K2, VGPR1=K3.
    int  row   = lane & 15;
    bool lo    = lane < 16;
    int  rc    = (row < 3) ? row : 0;
    float mrow = (row < 3) ? 1.0f : 0.0f;
    float a0 = (lo ? T[rc * 4 + 0] : T[rc * 4 + 2]) * mrow;
    float a1 = (lo ? T[rc * 4 + 1] : T[rc * 4 + 3]) * mrow;
    v2f A = {a0, a1};
    v8f c = {};

    // Two WMMAs per wave: group g covers pixels owned by lanes g*16..g*16+15.
    int waveBase = pBase + (threadIdx.x & ~31);
#pragma unroll
    for (int g = 0; g < 2; ++g) {
        int src = g * 16 + (lane & 15);                   // pixel-owner lane
        float sx = __shfl(xc, src, 32);
        float sy = __shfl(yc, src, 32);
        float sz = __shfl(zc, src, 32);
        // B 4x16 layout assumed symmetric to A: lane = column N;
        // lanes 0-15: VGPR0=K0(x), VGPR1=K1(y); lanes 16-31: VGPR0=K2(z), VGPR1=K3(1).
        float b0 = lo ? sx : sz;
        float b1 = lo ? sy : 1.0f;
        v2f Bv = {b0, b1};
        // D[i][n] = sum_k T[i][k] * depth_cam[k][n]   (exact f32)
        v8f dres = __builtin_amdgcn_wmma_f32_16x16x4_f32(
            false, A, false, Bv, (short)0, c, false, false);
        if (lo) {                                         // rows 0..2 live in lanes 0-15
            int pp = waveBase + g * 16 + lane;
            float* o = pc + ((size_t)b * HW_ + pp) * NF;
            o[0] = dres[0];
            o[1] = dres[1];
            o[2] = dres[2];
        }
    }

    // feature transpose: pc[b][p][3+c] = feat[b][c][p]  (coalesced reads)
    const float* fb = feat + ((size_t)b * C_) * HW_ + p;
    float* o = pc + ((size_t)b * HW_ + p) * NF + 3;
#pragma unroll
    for (int ch = 0; ch < C_; ++ch)
        o[ch] = fb[(size_t)ch * HW_];
}

__global__ __launch_bounds__(256)
void broadcast_kernel(const int* __restrict__ mask,
                      const float* __restrict__ pc,
                      float* __restrict__ out) {
    const unsigned perJB = HW_ * NF;                      // 1,075,200
    unsigned t    = blockIdx.x * 256u + threadIdx.x;
    unsigned base = t * 4u;                               // element index
    v4f o;
#pragma unroll
    for (int e = 0; e < 4; ++e) {
        unsigned idx = base + e;
        unsigned jb  = idx / perJB;                       // j*B + b
        unsigned rem = idx - jb * perJB;
        unsigned pp  = rem / NF;
        unsigned k   = rem - pp * NF;
        unsigned bb  = jb % B_;
        int m = mask[jb * HW_ + pp];                      // RT: stays in L2
        o[e] = m ? pc[((size_t)bb * HW_ + pp) * NF + k] : 0.0f;  // RT: pc L2-resident
    }
    // 258 MB write-once stream: non-temporal so it doesn't evict pc/mask from L2
    __builtin_nontemporal_store(o, reinterpret_cast<v4f*>(out + base));
}

extern "C" void kernel_launch(void* const* d_in, const int* in_sizes, int n_in,
                              void* d_out, int out_size, void* d_ws, size_t ws_size,
                              hipStream_t stream) {
    (void)in_sizes; (void)n_in; (void)out_size; (void)ws_size;
    const float* depth = (const float*)d_in[0];   // [B,1,H,W]
    const int*   mask  = (const int*)  d_in[1];   // [J,B,H,W]
    const float* K     = (const float*)d_in[2];   // [B,3,3]
    const float* M     = (const float*)d_in[3];   // [B,4,4]
    const float* diff  = (const float*)d_in[4];   // [B,12]
    const float* trans = (const float*)d_in[5];   // [B,4,4]
    const float* feat  = (const float*)d_in[6];   // [B,C,H,W]
    float* out = (float*)d_out;

    float* consts = (float*)d_ws;                 // 64 floats
    float* pc     = consts + 64;                  // B*HW*35 floats (~17.2 MB)

    setup_kernel<<<1, 32, 0, stream>>>(K, M, trans, consts);
    point_kernel<<<B_ * (HW_ / 256), 256, 0, stream>>>(depth, diff, feat, consts, pc);

    const unsigned total = (unsigned)J_ * B_ * HW_ * NF;  // 64,512,000 (div by 1024)
    broadcast_kernel<<<total / (4 * 256), 256, 0, stream>>>(mask, pc, out);
}